// MPS_17772574670987
// MI455X (gfx1250) — compile-verified
//
#include <hip/hip_runtime.h>
#include <hip/hip_bf16.h>

// ---------------------------------------------------------------------------
// MPS tensor-train forward on CDNA5 f32 WMMA (v_wmma_f32_16x16x4_f32).
//
// Per site:  tem_new[r,b] = sum_{l,i} A[l,i,r] * tem[l,b] * e_i[b]
//   M = r (bond out, 32 -> 2 M-tiles), N = b (16 batch rows / wave),
//   K = (l,i), 64 total -> 16 WMMA K-steps of 4.
//
// K-ordering is chosen so the B fragment needs NO cross-lane data movement:
//   step s uses l0 = (s<8 ? s : s+8); K=4s->(l0,0), 4s+1->(l0,1),
//   4s+2->(l0+8,0), 4s+3->(l0+8,1).
//   The D-tile register told[s&7] already holds tem[l0,b] in lanes 0-15 and
//   tem[l0+8,b] in lanes 16-31 -> B frag = {told*e0, told*e1} directly.
// The matching permutation of A is baked into the packed-weight layout, with
// both M-tiles' fragments interleaved so each lane loads one b128 per K-step.
// ---------------------------------------------------------------------------

typedef float v8f __attribute__((ext_vector_type(8)));
typedef float v4f __attribute__((ext_vector_type(4)));
typedef float v2f __attribute__((ext_vector_type(2)));

// Pack one site's raw parameter tensor into fragment order:
//   dst[((site*16 + s)*32 + lane)*4 + m*2 + i] = A[l, i, r]
//   with l = (s<8 ? s : s+8) + 8*(lane>=16), r = 16*m + (lane&15);
//   zero outside the site's true (L,R).
// kind: 0 = first site (2,2) 'ir', 1 = interior (L,2,R), 2 = last site (2,2) 'li'
__global__ __launch_bounds__(256) void mps_pack_site(
    const float* __restrict__ src, float* __restrict__ dst,
    int site, int L, int R, int kind)
{
    int t = blockIdx.x * blockDim.x + threadIdx.x;   // 0..2047
    if (t >= 2048) return;
    int i    = t & 1;
    int m    = (t >> 1) & 1;
    int lane = (t >> 2) & 31;
    int s    = (t >> 7) & 15;

    int l0 = (s < 8) ? s : (s + 8);
    int l  = l0 + ((lane >> 4) << 3);     // +8 for lanes 16..31
    int r  = 16 * m + (lane & 15);

    float v = 0.0f;
    if (kind == 0) {                 // params[0] shape (2,2): A[0,i,r] = P[i,r]
        if (l == 0 && r < 2) v = src[i * 2 + r];
    } else if (kind == 2) {          // params[N-1] shape (2,2): A[l,i,0] = P[l,i]
        if (l < 2 && r == 0) v = src[l * 2 + i];
    } else {                         // interior (L,2,R): A[l,i,r] = P[l,i,r]
        if (l < L && r < R) v = src[(l * 2 + i) * R + r];
    }
    dst[(((size_t)site * 16 + s) * 32 + lane) * 4 + m * 2 + i] = v;
}

// Chain kernel: one wave owns 16 batch rows for the full 64-site chain.
__global__ __launch_bounds__(256) void mps_chain(
    const float* __restrict__ x,     // [4096][64] spins in {-1,+1}
    const float* __restrict__ wpk,   // packed weights [64][16][32][4]
    float* __restrict__ out)         // [4096]
{
    const int lane  = threadIdx.x & 31;
    const int wave  = threadIdx.x >> 5;
    const int bbase = (blockIdx.x * 8 + wave) * 16;
    const int b     = bbase + (lane & 15);

    // Pack the 64 per-site embedding bits for this lane's batch row.
    // bit=1 <=> x == -1 <=> e = (0.01, 0.99); else (0.99, 0.01).
    unsigned long long mask = 0ull;
    const float4* x4 = (const float4*)(x + (size_t)b * 64);
    #pragma unroll
    for (int q = 0; q < 16; ++q) {
        float4 v = x4[q];
        unsigned long long mq = 0ull;
        mq |= (v.x < 0.0f) ? 1ull : 0ull;
        mq |= (v.y < 0.0f) ? 2ull : 0ull;
        mq |= (v.z < 0.0f) ? 4ull : 0ull;
        mq |= (v.w < 0.0f) ? 8ull : 0ull;
        mask |= mq << (q * 4);
    }

    // tem[l,b] in D-tile layout: told0 covers l=0..15, told1 covers l=16..31.
    // VGPR j: lanes 0-15 -> l=j (b=lane), lanes 16-31 -> l=j+8 (b=lane-16).
    // Initial state: tem[l,b] = (l == 0).
    v8f told0 = {};
    v8f told1 = {};
    told0[0] = (lane < 16) ? 1.0f : 0.0f;

    for (int site = 0; site < 64; ++site) {
        const float* wsite = wpk + (size_t)site * 2048;
        if (site < 63) __builtin_prefetch(wsite + 2048, 0, 0);  // next site

        const unsigned bit = (unsigned)((mask >> site) & 1ull);
        const float e0 = bit ? 0.01f : 0.99f;
        const float e1 = bit ? 0.99f : 0.01f;

        // Four independent accumulation chains (2 M-tiles x even/odd K-step)
        // -> dependent-WMMA chain length 8 instead of 16.
        v8f acc0a = {}, acc0b = {};
        v8f acc1a = {}, acc1b = {};

        #pragma unroll
        for (int s = 0; s < 16; ++s) {
            // B fragment: told[s&7] holds tem[l0,b] (lanes 0-15) and
            // tem[l0+8,b] (lanes 16-31) -- exactly the two K-rows needed.
            const v8f& tt = (s >> 3) ? told1 : told0;
            const float g = tt[s & 7];
            v2f bf;
            bf.x = g * e0;
            bf.y = g * e1;

            // Both M-tiles' A fragments in one coalesced b128 per lane.
            v4f a = *(const v4f*)(wsite + (s * 32 + lane) * 4);
            v2f a0; a0.x = a.x; a0.y = a.y;
            v2f a1; a1.x = a.z; a1.y = a.w;

            if (s & 1) {
                acc0b = __builtin_amdgcn_wmma_f32_16x16x4_f32(
                    false, a0, false, bf, (short)0, acc0b, false, false);
                acc1b = __builtin_amdgcn_wmma_f32_16x16x4_f32(
                    false, a1, false, bf, (short)0, acc1b, false, false);
            } else {
                acc0a = __builtin_amdgcn_wmma_f32_16x16x4_f32(
                    false, a0, false, bf, (short)0, acc0a, false, false);
                acc1a = __builtin_amdgcn_wmma_f32_16x16x4_f32(
                    false, a1, false, bf, (short)0, acc1a, false, false);
            }
        }

        told0 = acc0a + acc0b;
        told1 = acc1a + acc1b;
    }

    // Final site has r-dim 1 (padded): result is tem[0,b] = told0[0], lanes 0-15.
    if (lane < 16) out[b] = told0[0];
}

// ---------------------------------------------------------------------------

static inline int cap32(int e) { return (e >= 5) ? 32 : (1 << e); }

extern "C" void kernel_launch(void* const* d_in, const int* in_sizes, int n_in,
                              void* d_out, int out_size, void* d_ws, size_t ws_size,
                              hipStream_t stream) {
    const int N = 64;
    const float* x = (const float*)d_in[0];
    float* wpk = (float*)d_ws;   // 64 * 2048 floats = 512 KB

    // params may arrive as 64 separate pointers (d_in[1..64]) or as one
    // concatenated flat buffer (d_in[1]); handle both.
    const bool separate = (n_in >= 1 + N);
    size_t off = 0;
    for (int site = 0; site < N; ++site) {
        int kind, L = 0, R = 0, elems;
        if (site == 0)          { kind = 0; elems = 4; }
        else if (site == N - 1) { kind = 2; elems = 4; }
        else {
            kind = 1;
            if (site < N / 2) { L = cap32(site);         R = cap32(site + 1); }
            else              { R = cap32(N - 1 - site); L = cap32(N - site); }
            elems = L * 2 * R;
        }
        const float* p = separate ? (const float*)d_in[1 + site]
                                  : ((const float*)d_in[1]) + off;
        off += (size_t)elems;
        mps_pack_site<<<8, 256, 0, stream>>>(p, wpk, site, L, R, kind);
    }

    // 4096 batch rows / (16 per wave * 8 waves per block) = 32 blocks
    mps_chain<<<32, 256, 0, stream>>>(x, wpk, (float*)d_out);
}